// Net_65798898975283
// MI455X (gfx1250) — compile-verified
//
#include <hip/hip_runtime.h>

typedef float v2f __attribute__((ext_vector_type(2)));
typedef float v8f __attribute__((ext_vector_type(8)));

#define NNODES 20000
#define NEDGES 320000

// ---------------------------------------------------------------------------
// Edge scatter-add: agg[dst] += feat[src].  float4 gathers (global_load_b128)
// + native fp32 atomics (global_atomic_add_f32 via unsafeAtomicAdd).
// 20 MB accumulator lives in the 192 MB L2, so atomics stay on-chip.
// ---------------------------------------------------------------------------
__global__ __launch_bounds__(256) void k_scatter_add(
    const float* __restrict__ feat, const int* __restrict__ ei,
    float* __restrict__ agg, int n_edges, int D)
{
    const int chunks = D >> 2;  // float4 chunks per row
    long long tid = (long long)blockIdx.x * blockDim.x + threadIdx.x;
    long long total = (long long)n_edges * chunks;
    if (tid >= total) return;
    int e = (int)(tid / chunks);
    int c = (int)(tid - (long long)e * chunks);
    int s = ei[e];             // src row of edge_index
    int d = ei[n_edges + e];   // dst row of edge_index
    const float4 v = reinterpret_cast<const float4*>(feat + (size_t)s * D)[c];
    float* out = agg + (size_t)d * D + (c << 2);
    unsafeAtomicAdd(out + 0, v.x);
    unsafeAtomicAdd(out + 1, v.y);
    unsafeAtomicAdd(out + 2, v.z);
    unsafeAtomicAdd(out + 3, v.w);
}

// ---------------------------------------------------------------------------
// Strip-mined WMMA GEMM: C = act(A0@B0 [+ A1@B1] + bias), fp32 exact via
// V_WMMA_F32_16X16X4_F32.
//  - One wave owns a 16-row x N-col strip: NT=N/16 accumulator tiles, so a
//    single A fragment (global_load_b64) feeds NT WMMAs.
//  - Weights staged per 64-row K-chunk into LDS in K-PAIR-INTERLEAVED layout:
//    element (k, col) -> pair index (k/2)*N + col, slot (k&1).  A lane's B
//    fragment (B[k+2hi][col], B[k+2hi+1][col]) is then one contiguous 8-byte
//    v2f -> single ds_load_b64 straight into an even VGPR pair (no repack
//    movs before the WMMA).
//  - Staging: 2 global float4 loads (rows k, k+1) -> interleave in regs ->
//    2 ds_store_b128.
//  - 8 waves / 256-thread block -> 128 rows of M per block.  Inactive tail
//    waves still run the cooperative stage + barriers (wave-uniform guard,
//    EXEC all-ones for every WMMA).
// Fragment layouts per CDNA5 ISA:
//  A 16x4: lane<16 -> (K0,K1) of row M=lane; lane>=16 -> (K2,K3).
//  B 4x16: lane<16 -> col N=lane&15 rows (K0,K1); lane>=16 -> (K2,K3).
//  C/D: VGPR i -> M=i (lanes 0-15), M=i+8 (lanes 16-31), N=lane&15.
// ---------------------------------------------------------------------------
template <int NT, bool DUAL>
__global__ __launch_bounds__(256) void k_gemm_strip(
    const float* __restrict__ A0, const float* __restrict__ B0,
    const float* __restrict__ A1, const float* __restrict__ B1,
    const float* __restrict__ bias, float* __restrict__ C,
    int M, int K, int relu)
{
    constexpr int N = NT * 16;
    constexpr int BK = 64;                       // K-chunk staged in LDS
    __shared__ __align__(16) float Bs0[BK * N];  // <= 32 KB, pair-interleaved
    __shared__ __align__(16) float Bs1[DUAL ? (BK * N) : 4];

    const int lane = threadIdx.x & 31;
    const int wave = threadIdx.x >> 5;
    const int tm = blockIdx.x * 8 + wave;        // 16-row tile index in M
    const bool active = (tm * 16 < M);           // wave-uniform guard
    const int rh = lane & 15;
    const int hi = lane >> 4;

    v8f acc[NT];
#pragma unroll
    for (int t = 0; t < NT; ++t) acc[t] = v8f{};

    const float* ap0 = A0 + (size_t)(tm * 16 + rh) * K + 2 * hi;
    const float* ap1 = DUAL ? (A1 + (size_t)(tm * 16 + rh) * K + 2 * hi) : nullptr;

    for (int kc = 0; kc < K; kc += BK) {
        __syncthreads();
        // Cooperative stage with K-pair interleave:
        // iteration idx -> pairRow = idx/(N/4), col = (idx%(N/4))*4
        // loads rows k=2*pairRow, k+1 (float4 each), stores 2 x ds_store_b128.
        {
            constexpr int ITERS = (BK / 2) * (N / 4);
            for (int idx = threadIdx.x; idx < ITERS; idx += 256) {
                const int pr = idx / (N / 4);
                const int c = (idx - pr * (N / 4)) * 4;
                const int k = 2 * pr;
                float4 g0 = *(const float4*)(B0 + (size_t)(kc + k) * N + c);
                float4 g1 = *(const float4*)(B0 + (size_t)(kc + k + 1) * N + c);
                float4 lo = {g0.x, g1.x, g0.y, g1.y};
                float4 hi4 = {g0.z, g1.z, g0.w, g1.w};
                float* dst = &Bs0[(pr * N + c) * 2];
                *(float4*)(dst) = lo;
                *(float4*)(dst + 4) = hi4;
                if constexpr (DUAL) {
                    float4 h0 = *(const float4*)(B1 + (size_t)(kc + k) * N + c);
                    float4 h1 = *(const float4*)(B1 + (size_t)(kc + k + 1) * N + c);
                    float4 l2 = {h0.x, h1.x, h0.y, h1.y};
                    float4 h2 = {h0.z, h1.z, h0.w, h1.w};
                    float* dst1 = &Bs1[(pr * N + c) * 2];
                    *(float4*)(dst1) = l2;
                    *(float4*)(dst1 + 4) = h2;
                }
            }
        }
        __syncthreads();
        if (!active) continue;

        const v2f* Bp0 = (const v2f*)Bs0;
        const v2f* Bp1 = (const v2f*)Bs1;
        for (int k = 0; k < BK; k += 4) {
            const int pr = (k >> 1) + hi;        // pair row for this lane half
            v2f a;
            a.x = ap0[kc + k];
            a.y = ap0[kc + k + 1];
#pragma unroll
            for (int t = 0; t < NT; ++t) {
                v2f b = Bp0[pr * N + t * 16 + rh];   // one ds_load_b64
                acc[t] = __builtin_amdgcn_wmma_f32_16x16x4_f32(
                    false, a, false, b, (short)0, acc[t], false, false);
            }
            if constexpr (DUAL) {
                v2f a1;
                a1.x = ap1[kc + k];
                a1.y = ap1[kc + k + 1];
#pragma unroll
                for (int t = 0; t < NT; ++t) {
                    v2f b = Bp1[pr * N + t * 16 + rh];
                    acc[t] = __builtin_amdgcn_wmma_f32_16x16x4_f32(
                        false, a1, false, b, (short)0, acc[t], false, false);
                }
            }
        }
    }

    if (active) {
#pragma unroll
        for (int t = 0; t < NT; ++t) {
            const int col = t * 16 + rh;
            const float bv = bias[col];
#pragma unroll
            for (int i = 0; i < 8; ++i) {
                int r = tm * 16 + i + 8 * hi;
                float v = acc[t][i] + bv;
                if (relu) v = fmaxf(v, 0.0f);
                C[(size_t)r * N + col] = v;
            }
        }
    }
}

// ---------------------------------------------------------------------------
// Head: out = log_softmax(h @ lin3_w + lin3_b).  64->10, one thread per node.
// ---------------------------------------------------------------------------
__global__ __launch_bounds__(256) void k_head(
    const float* __restrict__ h, const float* __restrict__ W,
    const float* __restrict__ b, float* __restrict__ out, int M)
{
    int n = blockIdx.x * blockDim.x + threadIdx.x;
    if (n >= M) return;
    const float* hr = h + (size_t)n * 64;
    float z[10];
#pragma unroll
    for (int c = 0; c < 10; ++c) z[c] = b[c];
    for (int k = 0; k < 64; ++k) {
        float hv = hr[k];
        const float* wr = W + k * 10;
#pragma unroll
        for (int c = 0; c < 10; ++c) z[c] = fmaf(hv, wr[c], z[c]);
    }
    float mx = z[0];
#pragma unroll
    for (int c = 1; c < 10; ++c) mx = fmaxf(mx, z[c]);
    float s = 0.0f;
#pragma unroll
    for (int c = 0; c < 10; ++c) s += expf(z[c] - mx);
    float lse = mx + logf(s);
    float* o = out + (size_t)n * 10;
#pragma unroll
    for (int c = 0; c < 10; ++c) o[c] = z[c] - lse;
}

// ---------------------------------------------------------------------------
extern "C" void kernel_launch(void* const* d_in, const int* in_sizes, int n_in,
                              void* d_out, int out_size, void* d_ws, size_t ws_size,
                              hipStream_t stream) {
    const float* x       = (const float*)d_in[0];
    const int*   ei      = (const int*)d_in[1];   // [2, NEDGES]: src row then dst row
    const float* W1_rel  = (const float*)d_in[2];
    const float* W1_root = (const float*)d_in[3];
    const float* b1      = (const float*)d_in[4];
    const float* W2_rel  = (const float*)d_in[5];
    const float* W2_root = (const float*)d_in[6];
    const float* b2      = (const float*)d_in[7];
    const float* lin1_w  = (const float*)d_in[8];
    const float* lin1_b  = (const float*)d_in[9];
    const float* lin2_w  = (const float*)d_in[10];
    const float* lin2_b  = (const float*)d_in[11];
    const float* lin3_w  = (const float*)d_in[12];
    const float* lin3_b  = (const float*)d_in[13];

    // Workspace layout (floats), total ~61.5 MB
    float* agg1 = (float*)d_ws;                       // 20000*256
    float* h1   = agg1 + (size_t)NNODES * 256;        // 20000*128
    float* agg2 = h1   + (size_t)NNODES * 128;        // 20000*128
    float* h2   = agg2 + (size_t)NNODES * 128;        // 20000*64
    float* m1   = h2   + (size_t)NNODES * 64;         // 20000*128
    float* m2   = m1   + (size_t)NNODES * 128;        // 20000*64

    hipMemsetAsync(agg1, 0, (size_t)NNODES * 256 * sizeof(float), stream);
    hipMemsetAsync(agg2, 0, (size_t)NNODES * 128 * sizeof(float), stream);

    const int mblocks = ((NNODES / 16) + 7) / 8;      // 157 blocks, 8 waves each

    // GraphConv 1: agg1 = scatter(x); h1 = relu(agg1@W1_rel + x@W1_root + b1)
    {
        long long total = (long long)NEDGES * (256 / 4);
        int blocks = (int)((total + 255) / 256);
        k_scatter_add<<<blocks, 256, 0, stream>>>(x, ei, agg1, NEDGES, 256);
    }
    k_gemm_strip<8, true><<<mblocks, 256, 0, stream>>>(
        agg1, W1_rel, x, W1_root, b1, h1, NNODES, 256, 1);

    // GraphConv 2: agg2 = scatter(h1); h2 = agg2@W2_rel + h1@W2_root + b2
    {
        long long total = (long long)NEDGES * (128 / 4);
        int blocks = (int)((total + 255) / 256);
        k_scatter_add<<<blocks, 256, 0, stream>>>(h1, ei, agg2, NEDGES, 128);
    }
    k_gemm_strip<4, true><<<mblocks, 256, 0, stream>>>(
        agg2, W2_rel, h1, W2_root, b2, h2, NNODES, 128, 0);

    // MLP head
    k_gemm_strip<8, false><<<mblocks, 256, 0, stream>>>(
        h2, lin1_w, nullptr, nullptr, lin1_b, m1, NNODES, 64, 1);
    k_gemm_strip<4, false><<<mblocks, 256, 0, stream>>>(
        m1, lin2_w, nullptr, nullptr, lin2_b, m2, NNODES, 128, 1);
    k_head<<<(NNODES + 255) / 256, 256, 0, stream>>>(
        m2, lin3_w, lin3_b, (float*)d_out, NNODES);
}